// Decoder_5171140624512
// MI455X (gfx1250) — compile-verified
//
#include <hip/hip_runtime.h>
#include <math.h>

typedef __attribute__((ext_vector_type(2))) float v2f;
typedef __attribute__((ext_vector_type(8))) float v8f;

constexpr int B = 16, T = 256, D = 512, H = 512, V = 32000;
constexpr int G4H = 4 * H;     // 2048
constexpr int BH  = B * H;     // 8192

__device__ __forceinline__ float sigmoidf_(float x) {
  return 1.0f / (1.0f + __expf(-x));
}

__device__ __forceinline__ v8f wmma4(v2f a, v2f b, v8f acc) {
  return __builtin_amdgcn_wmma_f32_16x16x4_f32(false, a, false, b,
                                               (short)0, acc, false, false);
}

// acc += A[16 x K] * W[K x 16]  (A row stride lda, W row stride ldw, W cols start at n0)
// A frag (32b 16x4): lane%16 = M, VGPR j holds K = k0 + j + 2*(lane/16)
// B frag (4x16):     lane%16 = N, VGPR j holds K = k0 + j + 2*(lane/16)
__device__ __forceinline__ v8f wmma_accum_f32(v8f acc,
                                              const float* __restrict__ A, int lda,
                                              const float* __restrict__ W, int ldw,
                                              int n0, int K, int lane) {
  const int m  = lane & 15;
  const int hi = lane >> 4;          // 0 or 1
  const int n  = n0 + m;             // B-matrix column for this lane
  for (int k0 = 0; k0 < K; k0 += 4) {
    const int ka = k0 + 2 * hi;
    v2f a, b;
    a.x = A[m * lda + ka];
    a.y = A[m * lda + ka + 1];
    b.x = W[(size_t)ka * ldw + n];
    b.y = W[(size_t)(ka + 1) * ldw + n];
    acc = wmma4(a, b, acc);
  }
  return acc;
}

// One LSTM timestep for one layer. Grid: 32 blocks (H/16 column tiles), 128 threads
// (4 waves = gates i,f,g,o). z = X@Wx + h_in@Wh computed with f32 WMMA; gates fused via LDS.
__global__ __launch_bounds__(128) void lstm_step_kernel(
    const float* __restrict__ X, int ldx, int K,
    const float* __restrict__ Wx, const float* __restrict__ Wh,
    const float* __restrict__ bias,
    const float* __restrict__ h_in, float* __restrict__ h_out,
    float* __restrict__ c, float* __restrict__ yout, int t) {
  __shared__ float zs[4][16][16];
  const int nt   = blockIdx.x;          // column tile within H
  const int gate = threadIdx.x >> 5;    // 0..3
  const int lane = threadIdx.x & 31;
  const int n0   = gate * H + nt * 16;  // column base within [0, 4H)

  v8f acc = {};
  acc = wmma_accum_f32(acc, X,    ldx, Wx, G4H, n0, K, lane);
  acc = wmma_accum_f32(acc, h_in, H,   Wh, G4H, n0, H, lane);

  // C layout: VGPR r, lanes 0-15 -> M=r, lanes 16-31 -> M=r+8; N = lane%16
  const int hi  = lane >> 4;
  const int col = lane & 15;
#pragma unroll
  for (int r = 0; r < 8; ++r)
    zs[gate][r + 8 * hi][col] = acc[r];
  __syncthreads();

  for (int idx = threadIdx.x; idx < 256; idx += 128) {
    const int m = idx >> 4;
    const int j = idx & 15;
    const int n = nt * 16 + j;
    const float zi = zs[0][m][j] + bias[0 * H + n];
    const float zf = zs[1][m][j] + bias[1 * H + n];
    const float zg = zs[2][m][j] + bias[2 * H + n];
    const float zo = zs[3][m][j] + bias[3 * H + n];
    const float ig = sigmoidf_(zi);
    const float fg = sigmoidf_(zf);
    const float gg = tanhf(zg);
    const float og = sigmoidf_(zo);
    const float cn = fg * c[m * H + n] + ig * gg;
    const float hn = og * tanhf(cn);
    c[m * H + n]     = cn;
    h_out[m * H + n] = hn;
    if (yout) yout[((size_t)m * T + t) * H + n] = hn;   // y[B,T,H]
  }
}

// logits[BT x V] = Y[BT x H] @ W[H x V] + b.
// Register blocking: each wave computes 1 M-tile x 4 N-tiles, sharing one A
// fragment per k-step across 4 WMMAs (4x A reuse; GEMM is L2-bound, w is
// L2-resident at 65MB of the 192MB L2).
// Tasks: 256 (M) x 500 (N-groups of 64 cols) = 128000 = 16000 blocks * 8 waves.
__global__ __launch_bounds__(256) void logits_kernel(
    const float* __restrict__ Y, const float* __restrict__ W,
    const float* __restrict__ bout, float* __restrict__ logits) {
  const int wave = threadIdx.x >> 5;
  const int lane = threadIdx.x & 31;
  const int task = blockIdx.x * 8 + wave;
  const int mt   = task % (B * T / 16);   // 0..255
  const int ng   = task / (B * T / 16);   // 0..499
  const int n0   = ng * 64;

  const int m  = lane & 15;
  const int hi = lane >> 4;
  const float* __restrict__ A = Y + (size_t)mt * 16 * H;

  v8f acc0 = {}, acc1 = {}, acc2 = {}, acc3 = {};
  for (int k0 = 0; k0 < H; k0 += 4) {
    const int ka = k0 + 2 * hi;
    v2f a;
    a.x = A[m * H + ka];
    a.y = A[m * H + ka + 1];
    const float* __restrict__ Wr0 = W + (size_t)ka * V + n0 + m;
    const float* __restrict__ Wr1 = W + (size_t)(ka + 1) * V + n0 + m;
    v2f b0, b1, b2, b3;
    b0.x = Wr0[0];  b0.y = Wr1[0];
    b1.x = Wr0[16]; b1.y = Wr1[16];
    b2.x = Wr0[32]; b2.y = Wr1[32];
    b3.x = Wr0[48]; b3.y = Wr1[48];
    acc0 = wmma4(a, b0, acc0);
    acc1 = wmma4(a, b1, acc1);
    acc2 = wmma4(a, b2, acc2);
    acc3 = wmma4(a, b3, acc3);
  }

  const int colbase = n0 + m;
#pragma unroll
  for (int r = 0; r < 8; ++r) {
    const size_t row = (size_t)(mt * 16 + r + 8 * hi) * V;
    logits[row + colbase +  0] = acc0[r] + bout[colbase +  0];
    logits[row + colbase + 16] = acc1[r] + bout[colbase + 16];
    logits[row + colbase + 32] = acc2[r] + bout[colbase + 32];
    logits[row + colbase + 48] = acc3[r] + bout[colbase + 48];
  }
}

__global__ __launch_bounds__(256) void softmax_kernel(
    const float* __restrict__ logits, float* __restrict__ probs) {
  __shared__ float sdata[256];
  const int row = blockIdx.x;                 // 0..B*T-1
  const float* lrow = logits + (size_t)row * V;
  float* prow = probs + (size_t)row * V;

  float mx = -INFINITY;
  for (int i = threadIdx.x; i < V; i += 256) mx = fmaxf(mx, lrow[i]);
  sdata[threadIdx.x] = mx; __syncthreads();
  for (int s = 128; s > 0; s >>= 1) {
    if (threadIdx.x < s) sdata[threadIdx.x] = fmaxf(sdata[threadIdx.x], sdata[threadIdx.x + s]);
    __syncthreads();
  }
  mx = sdata[0]; __syncthreads();

  float sum = 0.0f;
  for (int i = threadIdx.x; i < V; i += 256) sum += __expf(lrow[i] - mx);
  sdata[threadIdx.x] = sum; __syncthreads();
  for (int s = 128; s > 0; s >>= 1) {
    if (threadIdx.x < s) sdata[threadIdx.x] += sdata[threadIdx.x + s];
    __syncthreads();
  }
  const float inv = 1.0f / sdata[0];
  for (int i = threadIdx.x; i < V; i += 256) prow[i] = __expf(lrow[i] - mx) * inv;
}

// Copy 4 state arrays of BH floats each: used both for init (inputs->ws) and final (ws->out).
__global__ __launch_bounds__(256) void copy4_kernel(
    const float* __restrict__ a, const float* __restrict__ b,
    const float* __restrict__ c, const float* __restrict__ d,
    float* __restrict__ da, float* __restrict__ db,
    float* __restrict__ dc, float* __restrict__ dd) {
  const int i = blockIdx.x * 256 + threadIdx.x;
  if (i < BH) { da[i] = a[i]; db[i] = b[i]; dc[i] = c[i]; dd[i] = d[i]; }
}

extern "C" void kernel_launch(void* const* d_in, const int* in_sizes, int n_in,
                              void* d_out, int out_size, void* d_ws, size_t ws_size,
                              hipStream_t stream) {
  (void)in_sizes; (void)n_in; (void)out_size; (void)ws_size;
  const float* x    = (const float*)d_in[0];
  const float* h0   = (const float*)d_in[1];
  const float* c0   = (const float*)d_in[2];
  const float* h1   = (const float*)d_in[3];
  const float* c1   = (const float*)d_in[4];
  const float* Wx0  = (const float*)d_in[5];
  const float* Wh0  = (const float*)d_in[6];
  const float* b0   = (const float*)d_in[7];
  const float* Wx1  = (const float*)d_in[8];
  const float* Wh1  = (const float*)d_in[9];
  const float* b1   = (const float*)d_in[10];
  const float* w    = (const float*)d_in[11];
  const float* bout = (const float*)d_in[12];

  // Workspace layout (floats)
  float* ws   = (float*)d_ws;
  float* y    = ws;                               // B*T*H
  float* h0b0 = ws + (size_t)B * T * H;           // ping
  float* h0b1 = h0b0 + BH;                        // pong
  float* c0w  = h0b1 + BH;
  float* h1b0 = c0w + BH;
  float* h1b1 = h1b0 + BH;
  float* c1w  = h1b1 + BH;

  // Output layout: probs | logits | a_h | a_c | b_h | b_c
  float* out    = (float*)d_out;
  float* probs  = out;
  float* logits = out + (size_t)B * T * V;
  float* states = out + 2 * (size_t)B * T * V;

  // init recurrent state from inputs (fresh every call -> deterministic)
  copy4_kernel<<<(BH + 255) / 256, 256, 0, stream>>>(h0, c0, h1, c1, h0b0, c0w, h1b0, c1w);

  float* h0buf[2] = {h0b0, h0b1};
  float* h1buf[2] = {h1b0, h1b1};
  for (int t = 0; t < T; ++t) {
    const float* hin0  = h0buf[t & 1];
    float*       hout0 = h0buf[(t + 1) & 1];
    lstm_step_kernel<<<H / 16, 128, 0, stream>>>(x + (size_t)t * D, T * D, D,
                                                 Wx0, Wh0, b0, hin0, hout0, c0w,
                                                 nullptr, t);
    const float* hin1  = h1buf[t & 1];
    float*       hout1 = h1buf[(t + 1) & 1];
    lstm_step_kernel<<<H / 16, 128, 0, stream>>>(hout0, H, H,
                                                 Wx1, Wh1, b1, hin1, hout1, c1w,
                                                 y, t);
  }
  // T even -> final states live in buffer index 0
  logits_kernel<<<(B * T / 16) * (V / 64) / 8, 256, 0, stream>>>(y, w, bout, logits);
  softmax_kernel<<<B * T, 256, 0, stream>>>(logits, probs);
  copy4_kernel<<<(BH + 255) / 256, 256, 0, stream>>>(
      h0buf[0], c0w, h1buf[0], c1w,
      states, states + BH, states + 2 * BH, states + 3 * BH);
}